// PatchFFT_14405320311466
// MI455X (gfx1250) — compile-verified
//
#include <hip/hip_runtime.h>
#include <math.h>

// MI455X / gfx1250, wave32. fp32 pipeline built on V_WMMA_F32_16X16X4_F32.
//
// Pipeline:
//   1) gemm_nt_bias: h[M,768] = x[M,768] . We[768,768]^T + be      (M = 25088)
//   2) fft_filter  : per 16x16 patch (75264 of them, in-place on h):
//        U = F h ; V = U F ; Z = V .* filt ; W = G Z ; R = W G ; a = |R|
//      with F = C - iS, G = (C + iS)/16  (C,S symmetric 16x16 DFT cos/sin)
//      -> every matmul is 4 chained v_wmma_f32_16x16x4_f32 ops.
//      Patch data is bulk-DMA'd into LDS with the Tensor Data Mover
//      (tensor_load_to_lds, tracked on TENSORcnt): one 8KB contiguous tile
//      per workgroup instead of 64 scattered global_load_b32.
//   3) gemm_nt_bias: out[M,768] = a[M,768] . Wd[768,768]^T + bd
//
// d_ws requirement: M*768*4 = 77,070,336 bytes for the intermediate h/a buffer.

typedef __attribute__((ext_vector_type(2))) float v2f;
typedef __attribute__((ext_vector_type(8))) float v8f;
typedef __attribute__((ext_vector_type(4))) unsigned int v4u;
typedef __attribute__((ext_vector_type(8))) int v8i;
typedef __attribute__((ext_vector_type(4))) int v4i;

#if defined(__has_builtin)
#if __has_builtin(__builtin_amdgcn_tensor_load_to_lds) && \
    __has_builtin(__builtin_amdgcn_s_wait_tensorcnt)
#define HAVE_TDM 1
#endif
#endif

static __device__ __forceinline__ v8f wmma_f32_k4(v2f a, v2f b, v8f c) {
  // D(16x16,f32) = A(16x4,f32) * B(4x16,f32) + C
  return __builtin_amdgcn_wmma_f32_16x16x4_f32(
      /*neg_a=*/false, a, /*neg_b=*/false, b,
      /*c_mod=*/(short)0, c, /*reuse_a=*/false, /*reuse_b=*/false);
}

// ---------------------------------------------------------------------------
// GEMM:  C[M,N] = A[M,K] * B[N,K]^T + bias[N]
// Block = 256 threads = 8 waves laid out 2(M) x 4(N); wave tile = 32x64.
// WG tile = 64(M) x 256(N). Grid = (M/64, N/256). K % 4 == 0.
// A-fragment (16x4 f32): lane(0-15)=row, v0/v1 hold K = 2*hi + {0,1}
// B-fragment (4x16 f32): lane(0-15)=col, v0/v1 hold K = 2*hi + {0,1}
// ---------------------------------------------------------------------------
__global__ __launch_bounds__(256) void gemm_nt_bias_kernel(
    const float* __restrict__ A, const float* __restrict__ B,
    const float* __restrict__ bias, float* __restrict__ C,
    int M, int N, int K) {
  const int lane = threadIdx.x & 31;
  const int wave = threadIdx.x >> 5;
  const int col  = lane & 15;
  const int hi   = lane >> 4;
  const int m0 = blockIdx.x * 64  + (wave >> 2) * 32;
  const int n0 = blockIdx.y * 256 + (wave & 3) * 64;

  const float* ap0 = A + (size_t)(m0 + col) * K + 2 * hi;
  const float* ap1 = ap0 + (size_t)16 * K;
  const float* bp0 = B + (size_t)(n0 + col) * K + 2 * hi;
  const float* bp1 = bp0 + (size_t)16 * K;
  const float* bp2 = bp0 + (size_t)32 * K;
  const float* bp3 = bp0 + (size_t)48 * K;

  const v8f vzero = {0.f,0.f,0.f,0.f,0.f,0.f,0.f,0.f};
  v8f acc[2][4];
  #pragma unroll
  for (int i = 0; i < 2; ++i)
    #pragma unroll
    for (int j = 0; j < 4; ++j) acc[i][j] = vzero;

  #pragma unroll 2
  for (int k = 0; k < K; k += 4) {
    // stream-ahead prefetch on the A operand into near caches (WGP scope)
    __builtin_prefetch(ap0 + k + 256, 0, 3);
    __builtin_prefetch(ap1 + k + 256, 0, 3);
    v2f a0 = *(const v2f*)(ap0 + k);
    v2f a1 = *(const v2f*)(ap1 + k);
    v2f b0 = *(const v2f*)(bp0 + k);
    v2f b1 = *(const v2f*)(bp1 + k);
    v2f b2 = *(const v2f*)(bp2 + k);
    v2f b3 = *(const v2f*)(bp3 + k);
    acc[0][0] = wmma_f32_k4(a0, b0, acc[0][0]);
    acc[0][1] = wmma_f32_k4(a0, b1, acc[0][1]);
    acc[0][2] = wmma_f32_k4(a0, b2, acc[0][2]);
    acc[0][3] = wmma_f32_k4(a0, b3, acc[0][3]);
    acc[1][0] = wmma_f32_k4(a1, b0, acc[1][0]);
    acc[1][1] = wmma_f32_k4(a1, b1, acc[1][1]);
    acc[1][2] = wmma_f32_k4(a1, b2, acc[1][2]);
    acc[1][3] = wmma_f32_k4(a1, b3, acc[1][3]);
  }

  // Epilogue: D layout is M = r + 8*hi, N = lane&15 (per 16x16 tile).
  #pragma unroll
  for (int mt = 0; mt < 2; ++mt)
    #pragma unroll
    for (int nt = 0; nt < 4; ++nt) {
      const int n = n0 + nt * 16 + col;
      const float bv = bias[n];
      #pragma unroll
      for (int r = 0; r < 8; ++r) {
        const int m = m0 + mt * 16 + r + 8 * hi;
        C[(size_t)m * N + n] = acc[mt][nt][r] + bv;
      }
    }
}

// ---------------------------------------------------------------------------
// Per-patch FFT -> complex filter -> IFFT -> abs, in place on H.
// One wave per 16x16 patch; 8 waves per block. 56 WMMAs per patch.
// The block's 8 patches (8KB contiguous) are staged into LDS by the TDM.
// LDS (stride-17 slices) also converts WMMA D-layout back to A-/B-layouts.
// ---------------------------------------------------------------------------
__global__ __launch_bounds__(256) void fft_filter_kernel(
    float* __restrict__ H, const float* __restrict__ Fre,
    const float* __restrict__ Fim) {
  __shared__ float sre[8][16 * 17];
  __shared__ float sim[8][16 * 17];
#if HAVE_TDM
  __shared__ float stage[8][256];
#endif

  const int lane = threadIdx.x & 31;
  const int wave = threadIdx.x >> 5;
  const int col  = lane & 15;   // column for B/D layouts, row for A layout
  const int hi   = lane >> 4;

  float* patch = H + ((size_t)blockIdx.x * 8 + wave) * 256;

#if HAVE_TDM
  // --- Tensor Data Mover: DMA this block's 8 patches (8192B) into LDS. ---
  if (wave == 0) {
    const unsigned long long gaddr =
        (unsigned long long)(uintptr_t)H + (unsigned long long)blockIdx.x * 8192ull;
    const unsigned int lds_addr = (unsigned int)(uintptr_t)(&stage[0][0]);
    const unsigned int nelem = 2048u;       // 8 patches * 256 f32
    // D# group 0: count=1(valid), lds_addr, 57b global byte address, type=2
    v4u g0;
    g0.x = 1u;
    g0.y = lds_addr;
    g0.z = (unsigned int)(gaddr & 0xffffffffull);
    g0.w = (unsigned int)((gaddr >> 32) & 0x01ffffffull) | (2u << 30);
    // D# group 1: data_size=4B, tensor_dim0 (32b @bit48), tensor_dim1=1,
    //             tile_dim0=2048 (@bit112), tile_dim1=1, strides.
    const unsigned int tdim0 = nelem;       // whole transfer is one row
    v8i g1;
    g1[0] = (int)(2u << 16);                          // data_size = 4 bytes
    g1[1] = (int)((tdim0 & 0xffffu) << 16);           // tensor_dim0 [15:0]
    g1[2] = (int)((tdim0 >> 16) | (1u << 16));        // dim0[31:16] | tensor_dim1=1
    g1[3] = (int)((nelem & 0xffffu) << 16);           // tile_dim0 = 2048
    g1[4] = 1;                                        // tile_dim1 = 1
    g1[5] = (int)nelem;                               // tensor_dim0_stride
    g1[6] = 0;
    g1[7] = 0;
    v4i gz4 = {0, 0, 0, 0};
    v8i gz8 = {0, 0, 0, 0, 0, 0, 0, 0};
    // clang-23 / therock-10.0 lane: 6-arg form (g0, g1, g2, g3, g4, cpol)
    __builtin_amdgcn_tensor_load_to_lds(g0, g1, gz4, gz4, gz8, 0);
    __builtin_amdgcn_s_wait_tensorcnt(0);
  }
  __syncthreads();
  const float* psrc = stage[wave];
#else
  const float* psrc = patch;
#endif

  // DFT constant fragments. C[j][k]=cos(2*pi*j*k/16), S[j][k]=sin(2*pi*j*k/16).
  // C,S symmetric  =>  A-layout and B-layout per-lane fragments are identical.
  v2f cf[4], sf[4], nsf[4], cg[4], sg[4], nsg[4];
  const float w = 0.39269908169872414f;  // 2*pi/16
  #pragma unroll
  for (int j = 0; j < 4; ++j) {
    const int k0 = 4 * j + 2 * hi;
    float s0, c0, s1, c1;
    __sincosf(w * (float)((col * k0) & 15), &s0, &c0);
    __sincosf(w * (float)((col * (k0 + 1)) & 15), &s1, &c1);
    cf[j].x = c0; cf[j].y = c1;
    sf[j].x = s0; sf[j].y = s1;
    nsf[j] = -sf[j];
    cg[j]  = cf[j] * 0.0625f;   // G = (C + iS)/16
    sg[j]  = sf[j] * 0.0625f;
    nsg[j] = -sg[j];
  }

  // Load h in B-operand layout: B[k][col] = patch[k*16+col].
  v2f hb[4];
  #pragma unroll
  for (int j = 0; j < 4; ++j) {
    const int k0 = 4 * j + 2 * hi;
    hb[j].x = psrc[k0 * 16 + col];
    hb[j].y = psrc[(k0 + 1) * 16 + col];
  }

  const v8f vzero = {0.f,0.f,0.f,0.f,0.f,0.f,0.f,0.f};

  // U = F*h : Ure = C*h, Uim = -S*h
  v8f ure = vzero, uim = vzero;
  #pragma unroll
  for (int j = 0; j < 4; ++j) {
    ure = wmma_f32_k4(cf[j],  hb[j], ure);
    uim = wmma_f32_k4(nsf[j], hb[j], uim);
  }

  float* wr = sre[wave];
  float* wi = sim[wave];

  // D-layout -> LDS
  #pragma unroll
  for (int r = 0; r < 8; ++r) {
    wr[(r + 8 * hi) * 17 + col] = ure[r];
    wi[(r + 8 * hi) * 17 + col] = uim[r];
  }
  __syncthreads();

  // U in A-operand layout: A[row][k] with row = col.
  v2f ar[4], ai[4];
  #pragma unroll
  for (int j = 0; j < 4; ++j) {
    const int k0 = 4 * j + 2 * hi;
    ar[j].x = wr[col * 17 + k0];  ar[j].y = wr[col * 17 + k0 + 1];
    ai[j].x = wi[col * 17 + k0];  ai[j].y = wi[col * 17 + k0 + 1];
  }

  // V = U*F : Vre = Ure*C + Uim*S ; Vim = Uim*C - Ure*S
  v8f vre = vzero, vim = vzero;
  #pragma unroll
  for (int j = 0; j < 4; ++j) {
    vre = wmma_f32_k4(ar[j], cf[j],  vre);
    vre = wmma_f32_k4(ai[j], sf[j],  vre);
    vim = wmma_f32_k4(ai[j], cf[j],  vim);
    vim = wmma_f32_k4(ar[j], nsf[j], vim);
  }

  // Z = V .* (Fre + i*Fim), elementwise in D layout.
  #pragma unroll
  for (int r = 0; r < 8; ++r) {
    const int rr = r + 8 * hi;
    const float fr = Fre[rr * 16 + col];
    const float fi = Fim[rr * 16 + col];
    const float zr = vre[r] * fr - vim[r] * fi;
    const float zi = vre[r] * fi + vim[r] * fr;
    vre[r] = zr; vim[r] = zi;
  }

  __syncthreads();  // before reusing LDS
  #pragma unroll
  for (int r = 0; r < 8; ++r) {
    wr[(r + 8 * hi) * 17 + col] = vre[r];
    wi[(r + 8 * hi) * 17 + col] = vim[r];
  }
  __syncthreads();

  // Z in B-operand layout.
  v2f zr[4], zi[4];
  #pragma unroll
  for (int j = 0; j < 4; ++j) {
    const int k0 = 4 * j + 2 * hi;
    zr[j].x = wr[k0 * 17 + col];  zr[j].y = wr[(k0 + 1) * 17 + col];
    zi[j].x = wi[k0 * 17 + col];  zi[j].y = wi[(k0 + 1) * 17 + col];
  }

  // W = G*Z : Wre = Cg*Zre - Sg*Zim ; Wim = Cg*Zim + Sg*Zre
  v8f yre = vzero, yim = vzero;
  #pragma unroll
  for (int j = 0; j < 4; ++j) {
    yre = wmma_f32_k4(cg[j],  zr[j], yre);
    yre = wmma_f32_k4(nsg[j], zi[j], yre);
    yim = wmma_f32_k4(cg[j],  zi[j], yim);
    yim = wmma_f32_k4(sg[j],  zr[j], yim);
  }

  __syncthreads();
  #pragma unroll
  for (int r = 0; r < 8; ++r) {
    wr[(r + 8 * hi) * 17 + col] = yre[r];
    wi[(r + 8 * hi) * 17 + col] = yim[r];
  }
  __syncthreads();

  // W in A-operand layout.
  v2f pr[4], pi[4];
  #pragma unroll
  for (int j = 0; j < 4; ++j) {
    const int k0 = 4 * j + 2 * hi;
    pr[j].x = wr[col * 17 + k0];  pr[j].y = wr[col * 17 + k0 + 1];
    pi[j].x = wi[col * 17 + k0];  pi[j].y = wi[col * 17 + k0 + 1];
  }

  // R = W*G : Rre = Wre*Cg - Wim*Sg ; Rim = Wim*Cg + Wre*Sg
  v8f rre = vzero, rim = vzero;
  #pragma unroll
  for (int j = 0; j < 4; ++j) {
    rre = wmma_f32_k4(pr[j], cg[j],  rre);
    rre = wmma_f32_k4(pi[j], nsg[j], rre);
    rim = wmma_f32_k4(pi[j], cg[j],  rim);
    rim = wmma_f32_k4(pr[j], sg[j],  rim);
  }

  // a = |R|, store in place (D layout scatter).
  #pragma unroll
  for (int r = 0; r < 8; ++r) {
    const int rr = r + 8 * hi;
    patch[rr * 16 + col] = sqrtf(rre[r] * rre[r] + rim[r] * rim[r]);
  }
}

// ---------------------------------------------------------------------------
extern "C" void kernel_launch(void* const* d_in, const int* in_sizes, int n_in,
                              void* d_out, int out_size, void* d_ws, size_t ws_size,
                              hipStream_t stream) {
  const float* x   = (const float*)d_in[0];  // [B,P,E]   = [128,196,768]
  const float* We  = (const float*)d_in[1];  // [C,E]     = [768,768]
  const float* be  = (const float*)d_in[2];  // [C]
  const float* fre = (const float*)d_in[3];  // [16,16]
  const float* fim = (const float*)d_in[4];  // [16,16]
  const float* Wd  = (const float*)d_in[5];  // [E,C]
  const float* bd  = (const float*)d_in[6];  // [E]
  float* out = (float*)d_out;

  const int E = 768;
  const int Cdim = in_sizes[1] / E;   // ch*ps*ps = 768
  const int M = in_sizes[0] / E;      // B*P = 25088

  float* h = (float*)d_ws;            // M * Cdim fp32 intermediate (77 MB)

  dim3 blk(256);

  // encode: h = x * We^T + be
  dim3 g1(M / 64, Cdim / 256);
  gemm_nt_bias_kernel<<<g1, blk, 0, stream>>>(x, We, be, h, M, Cdim, E);

  // per-patch FFT/filter/IFFT/abs (in place on h)
  const int npatch = M * (Cdim / 256);  // 75264 patches, 8 per block
  fft_filter_kernel<<<dim3(npatch / 8), blk, 0, stream>>>(h, fre, fim);

  // decode: out = a * Wd^T + bd
  gemm_nt_bias_kernel<<<g1, blk, 0, stream>>>(h, Wd, bd, out, M, E, Cdim);
}